// Style_dubber_model_loss_15_Emo_30623116820870
// MI455X (gfx1250) — compile-verified
//
#include <hip/hip_runtime.h>
#include <math.h>

// Problem constants (match reference)
#define B_      16
#define T_SRC_  512
#define T_MEL_  2048
#define T_LIP_  1024
#define N_MELS_ 80
#define D_SPK_  256
#define N_EMO_  8

// Attention tiling: 8 rows/block, 4-deep async ring buffer
#define ROWS_A   8
#define DEPTH_A  4
#define CHUNKS_A (T_SRC_ / ROWS_A)      // 64 chunks per batch
#define NUM_A_BLOCKS (B_ * CHUNKS_A)    // 1024 blocks

// Mel tiling
#define CHUNKS_M 80                     // 2048-float chunks per batch
#define MEL_PER_B (T_MEL_ * N_MELS_)    // 163840
#define MEL_CHUNK (MEL_PER_B / CHUNKS_M)// 2048 floats = 512 float4
#define NUM_M_BLOCKS (B_ * CHUNKS_M)    // 1280

__device__ __forceinline__ float wave_sum32(float v) {
#pragma unroll
  for (int m = 16; m > 0; m >>= 1) v += __shfl_xor(v, m, 32);
  return v;
}

// Reduce two values across a 256-thread block; result valid on thread 0.
__device__ __forceinline__ void block_sum2_256(float& a, float& b,
                                               float (*wred)[2]) {
  a = wave_sum32(a);
  b = wave_sum32(b);
  const int lane = threadIdx.x & 31;
  const int w = threadIdx.x >> 5;
  if (lane == 0) { wred[w][0] = a; wred[w][1] = b; }
  __syncthreads();
  if (threadIdx.x == 0) {
    float sa = 0.f, sb = 0.f;
#pragma unroll
    for (int i = 0; i < 8; ++i) { sa += wred[i][0]; sb += wred[i][1]; }
    a = sa; b = sb;
  }
  __syncthreads();
}

__device__ __forceinline__ float block_sum_256(float v, float* red) {
  int t = threadIdx.x;
  red[t] = v;
  __syncthreads();
  for (int s = 128; s > 0; s >>= 1) {
    if (t < s) red[t] += red[t + s];
    __syncthreads();
  }
  float r = red[0];
  __syncthreads();
  return r;
}

// low 32 bits of a flat LDS pointer == DS byte address on gfx1250
__device__ __forceinline__ unsigned lds_addr_of(const void* p) {
  return (unsigned)(unsigned long long)p;
}

// ---------------------------------------------------------------------------
// Kernel 1: diagonal-attention reduction over attn_scores [B, T_src, T_lip].
// 4-deep async global->LDS ring pipeline (ASYNCcnt / DScnt split counters).
// Each thread owns a disjoint 16B slice of each 4KB row, so per-wave
// s_wait_asynccnt is sufficient (no block barrier in the pipeline).
// ---------------------------------------------------------------------------
__global__ __launch_bounds__(256) void attn_diag_kernel(
    const float* __restrict__ attn, const int* __restrict__ x_len,
    const int* __restrict__ lip_len, float* __restrict__ part) {
  __shared__ float buf[DEPTH_A][T_LIP_];   // 16 KB ring (4 x 4 KB rows)
  __shared__ float wred[8][2];

  const int blk = blockIdx.x;
  const int b = blk / CHUNKS_A;
  const int c = blk % CHUNKS_A;
  const int t = threadIdx.x;
  const int ll0 = c * ROWS_A;

  const int xlI = x_len[b];
  const int lipI = lip_len[b];
  const float ks = (float)lipI / (float)xlI;   // attn_ks
  const float da = (float)lipI * 0.125f;       // lip_len / DIA (DIA = 8)

  const float* g = attn + ((size_t)b * T_SRC_ + (size_t)ll0) * T_LIP_ + t * 4;
  unsigned lds[DEPTH_A];
#pragma unroll
  for (int i = 0; i < DEPTH_A; ++i) lds[i] = lds_addr_of(&buf[i][t * 4]);

  // Prime the pipeline: rows 0..3 in flight (per-lane 16B async DMA to LDS)
#pragma unroll
  for (int i = 0; i < DEPTH_A; ++i) {
    asm volatile("global_load_async_to_lds_b128 %0, %1, off"
                 :: "v"(lds[i]), "v"(g + (size_t)i * T_LIP_) : "memory");
  }

  float num = 0.f, den = 0.f;
#pragma unroll
  for (int r = 0; r < ROWS_A; ++r) {
    // Async loads retire in order: wait <=N leaves the newest N outstanding.
    if (r < ROWS_A - 3)      asm volatile("s_wait_asynccnt 0x3" ::: "memory");
    else if (r == ROWS_A - 3) asm volatile("s_wait_asynccnt 0x2" ::: "memory");
    else if (r == ROWS_A - 2) asm volatile("s_wait_asynccnt 0x1" ::: "memory");
    else                      asm volatile("s_wait_asynccnt 0x0" ::: "memory");

    float4 v = *(const float4*)&buf[r & (DEPTH_A - 1)][t * 4];

    const int ll = ll0 + r;
    const bool vll = ll < xlI;
    const float center = ks * (float)ll;
    const float y1 = floorf(center + da);
    const float y2 = floorf(fmaxf(center - da, 0.f));

    const float vv[4] = {v.x, v.y, v.z, v.w};
#pragma unroll
    for (int j = 0; j < 4; ++j) {
      const int yy = t * 4 + j;
      const float a = (vll && (yy < lipI)) ? vv[j] : 0.f;
      den += a;
      const float fy = (float)yy;
      if (fy >= y2 && fy < y1) num += a;
    }

    if (r + DEPTH_A < ROWS_A) {
      // Our ds_load of this buffer must land before the async DMA may
      // overwrite it; then issue row r+4 into the same ring slot.
      asm volatile("s_wait_dscnt 0x0\n\t"
                   "global_load_async_to_lds_b128 %0, %1, off"
                   :: "v"(lds[r & (DEPTH_A - 1)]),
                      "v"(g + (size_t)(r + DEPTH_A) * T_LIP_)
                   : "memory");
    }
  }

  __syncthreads();
  block_sum2_256(num, den, wred);
  if (t == 0) {
    part[2 * blk]     = num;
    part[2 * blk + 1] = den;
  }
}

// ---------------------------------------------------------------------------
// Kernel 2: masked L1 sums for mel + postnet-mel predictions vs targets.
// Plain b128 streaming (bandwidth-bound). Writes (s_mel, s_post) per block.
// ---------------------------------------------------------------------------
__global__ __launch_bounds__(256) void mel_l1_kernel(
    const float* __restrict__ tgt, const float* __restrict__ pred,
    const float* __restrict__ post, const int* __restrict__ mel_lens,
    float* __restrict__ part) {
  __shared__ float wred[8][2];
  const int blk = blockIdx.x;
  const int b = blk / CHUNKS_M;
  const int c = blk % CHUNKS_M;
  const int ml = mel_lens[b];
  const size_t base = (size_t)b * MEL_PER_B + (size_t)c * MEL_CHUNK;

  float s1 = 0.f, s2 = 0.f;
#pragma unroll
  for (int k = 0; k < 2; ++k) {
    const int q = threadIdx.x + k * 256;   // float4 index in chunk, < 512
    const int e = c * MEL_CHUNK + q * 4;   // element index within batch slab
    const int trow = e / N_MELS_;          // 80 | 4 -> float4 never straddles t
    const float4 a  = *(const float4*)(tgt  + base + q * 4);
    const float4 p1 = *(const float4*)(pred + base + q * 4);
    const float4 p2 = *(const float4*)(post + base + q * 4);
    if (trow < ml) {
      s1 += fabsf(p1.x - a.x) + fabsf(p1.y - a.y) +
            fabsf(p1.z - a.z) + fabsf(p1.w - a.w);
      s2 += fabsf(p2.x - a.x) + fabsf(p2.y - a.y) +
            fabsf(p2.z - a.z) + fabsf(p2.w - a.w);
    }
  }
  block_sum2_256(s1, s2, wred);
  if (threadIdx.x == 0) {
    part[2 * blk]     = s1;
    part[2 * blk + 1] = s2;
  }
}

// ---------------------------------------------------------------------------
// Kernel 3: one block. Duration MSE, speaker cosine, emotion CE, combine all
// partials deterministically, write the 7 outputs in reference tuple order.
// ---------------------------------------------------------------------------
__global__ __launch_bounds__(256) void finalize_kernel(
    const float* __restrict__ part, const int* __restrict__ mel_lens,
    const int* __restrict__ dur_tgt, const float* __restrict__ log_dur_pred,
    const int* __restrict__ x_len, const float* __restrict__ spk_pred,
    const float* __restrict__ spk_emb, const float* __restrict__ emo_emb,
    const int* __restrict__ emo_id, float* __restrict__ out) {
  __shared__ float red[256];
  const int t = threadIdx.x;

  // ---- focus mean (per-batch num/den from attn partials) ----
  float focus = 0.f;
  if (t < B_) {
    float n = 0.f, d = 0.f;
    for (int c = 0; c < CHUNKS_A; ++c) {
      n += part[(t * CHUNKS_A + c) * 2];
      d += part[(t * CHUNKS_A + c) * 2 + 1];
    }
    focus = n / d;
  }
  const float focus_sum = block_sum_256(focus, red);

  // ---- mel L1 sums ----
  const float* mp = part + 2 * NUM_A_BLOCKS;
  float m1 = 0.f, m2 = 0.f;
  for (int i = t; i < NUM_M_BLOCKS; i += 256) {
    m1 += mp[2 * i];
    m2 += mp[2 * i + 1];
  }
  m1 = block_sum_256(m1, red);
  m2 = block_sum_256(m2, red);

  // ---- masked log-duration MSE ----
  float ssq = 0.f;
  for (int i = t; i < B_ * T_SRC_; i += 256) {
    const int b = i >> 9;
    const int tt = i & (T_SRC_ - 1);
    if (tt < x_len[b]) {
      const float d = log_dur_pred[i] - logf((float)dur_tgt[i] + 1.f);
      ssq += d * d;
    }
  }
  ssq = block_sum_256(ssq, red);

  // ---- speaker cosine ----
  float spk = 0.f;
  if (t < B_) {
    float dot = 0.f, na2 = 0.f, nb2 = 0.f;
    for (int k = 0; k < D_SPK_; ++k) {
      const float a = spk_pred[t * D_SPK_ + k];
      const float b = spk_emb[t * D_SPK_ + k];
      dot += a * b; na2 += a * a; nb2 += b * b;
    }
    const float na = fmaxf(sqrtf(na2), 1e-6f);
    const float nb = fmaxf(sqrtf(nb2), 1e-6f);
    spk = 1.f - dot / (na * nb);
  }
  spk = block_sum_256(spk, red);

  // ---- emotion cross-entropy ----
  float emo = 0.f;
  if (t < B_) {
    float mx = -3.0e38f;
    for (int k = 0; k < N_EMO_; ++k) mx = fmaxf(mx, emo_emb[t * N_EMO_ + k]);
    float se = 0.f;
    for (int k = 0; k < N_EMO_; ++k) se += expf(emo_emb[t * N_EMO_ + k] - mx);
    const float lse = mx + logf(se);
    emo = lse - emo_emb[t * N_EMO_ + emo_id[t]];
  }
  emo = block_sum_256(emo, red);

  if (t == 0) {
    float nmel = 0.f, nsrc = 0.f;
    for (int b = 0; b < B_; ++b) {
      nmel += (float)mel_lens[b];
      nsrc += (float)x_len[b];
    }
    nmel *= (float)N_MELS_;
    const float mel_loss  = m1 / nmel;
    const float post_loss = m2 / nmel;
    const float dur_loss  = ssq / nsrc;
    const float spk_loss  = spk / (float)B_;
    const float emo_loss  = emo / (float)B_;
    const float diag_loss = -logf(focus_sum / (float)B_);
    const float total = mel_loss + post_loss + dur_loss +
                        spk_loss + emo_loss + diag_loss;
    out[0] = total;
    out[1] = mel_loss;
    out[2] = post_loss;
    out[3] = spk_loss;
    out[4] = dur_loss;
    out[5] = diag_loss;
    out[6] = emo_loss;
  }
}

extern "C" void kernel_launch(void* const* d_in, const int* in_sizes, int n_in,
                              void* d_out, int out_size, void* d_ws, size_t ws_size,
                              hipStream_t stream) {
  // setup_inputs() order; bool mask arrays (11,12,13,19) deliberately unused —
  // masks are recomputed from the integer length arrays.
  const float* mel_tgt   = (const float*)d_in[2];
  const int*   mel_lens  = (const int*)  d_in[5];
  const int*   dur_tgt   = (const int*)  d_in[6];
  const float* mel_pred  = (const float*)d_in[8];
  const float* post_pred = (const float*)d_in[9];
  const float* log_dur   = (const float*)d_in[10];
  const float* spk_pred  = (const float*)d_in[14];
  const float* spk_emb   = (const float*)d_in[15];
  const float* attn      = (const float*)d_in[16];
  const int*   x_len     = (const int*)  d_in[17];
  const int*   lip_len   = (const int*)  d_in[18];
  const float* emo_emb   = (const float*)d_in[20];
  const int*   emo_id    = (const int*)  d_in[21];
  float* ws  = (float*)d_ws;
  float* out = (float*)d_out;

  attn_diag_kernel<<<NUM_A_BLOCKS, 256, 0, stream>>>(attn, x_len, lip_len, ws);
  mel_l1_kernel<<<NUM_M_BLOCKS, 256, 0, stream>>>(mel_tgt, mel_pred, post_pred,
                                                  mel_lens,
                                                  ws + 2 * NUM_A_BLOCKS);
  finalize_kernel<<<1, 256, 0, stream>>>(ws, mel_lens, dur_tgt, log_dur, x_len,
                                         spk_pred, spk_emb, emo_emb, emo_id,
                                         out);
}